// CombinedQLSTM_65481071397708
// MI455X (gfx1250) — compile-verified
//
#include <hip/hip_runtime.h>
#include <math.h>

// Problem dims (fixed by reference)
#define T_DIM 1024
#define B_DIM 128
#define D_DIM 256
#define H_DIM 256
#define ROWS  (T_DIM * B_DIM)   // 131072 rows of the [T*B, D] input
#define A_STR 260               // padded LDS row stride (floats): banks = 4*row + k, conflict-free
#define W_STR 260

typedef __attribute__((ext_vector_type(2))) float v2f;
typedef __attribute__((ext_vector_type(8))) float v8f;
typedef __attribute__((ext_vector_type(4))) unsigned int v4u;
typedef __attribute__((ext_vector_type(8))) int v8i;
typedef __attribute__((ext_vector_type(4))) int v4i;

__device__ __forceinline__ float fast_tanh(float x) {
#if __has_builtin(__builtin_amdgcn_tanhf)
    return __builtin_amdgcn_tanhf(x);   // gfx1250 v_tanh_f32
#else
    return tanhf(x);
#endif
}
__device__ __forceinline__ float fast_sigmoid(float x) {
    return 0.5f * fast_tanh(0.5f * x) + 0.5f;   // exact identity, 1 trans op
}

// ---------------------------------------------------------------------------
// Phase A: xdot[row][k] = sum_d in[row][d] * W[k][d]   (k = 0..3, d = 0..255)
// fp32 WMMA 16x16x4: per wave, M=16 rows, N=16 (4 used), K accumulated 4/step.
// A tile (32 rows x 1 KB) staged to LDS by the Tensor Data Mover with HW row
// padding (+16 B per 1 KB row) to keep the ds reads bank-conflict-free.
// ---------------------------------------------------------------------------
__global__ __launch_bounds__(64) void qlstm_xdot_wmma(const float* __restrict__ in,
                                                      const float* __restrict__ W,
                                                      float* __restrict__ xdot) {
    __shared__ __align__(16) float Ain[32 * A_STR];   // 33,280 B
    __shared__ __align__(16) float Wl[16 * W_STR];    // 16,640 B (rows 4..15 zero)
    const int tid     = threadIdx.x;
    const int rowBase = blockIdx.x * 32;

    // Stage B operand: W[0..3][0..255]; N columns 4..15 hard-zeroed in LDS so
    // the inner loop needs no mask VALU on the WMMA critical path.
    for (int idx = tid; idx < 16 * 256; idx += 64) {
        int n = idx >> 8, k = idx & 255;
        Wl[n * W_STR + k] = (n < 4) ? W[n * 512 + k] : 0.0f;
    }

#if __has_builtin(__builtin_amdgcn_tensor_load_to_lds) && __has_builtin(__builtin_amdgcn_s_wait_tensorcnt)
    // ---- TDM path: one 2-D tile DMA, global -> LDS, with hardware padding ----
    if (tid < 32) {                      // wave 0 only (EXEC ignored by TDM; branch is per-wave)
        const unsigned long long ga = (unsigned long long)(uintptr_t)(in + (size_t)rowBase * 256);
        const unsigned int ldsA = (unsigned int)(uintptr_t)&Ain[0];   // flat LDS addr low 32 = LDS offset

        // D# group 0: count=1 | lds_addr | global_addr[56:0] | type=2 ("image")
        v4u g0;
        g0.x = 1u;                                   // count=1, user descriptor
        g0.y = ldsA;                                 // lds_addr (bytes)
        g0.z = (unsigned int)(ga & 0xFFFFFFFFu);     // global_addr[31:0]
        g0.w = (unsigned int)((ga >> 32) & 0x1FFFFFFu) | (2u << 30);  // global_addr[56:32], type=2

        // D# group 1: data_size=2 (4 B), pad_enable, pad_interval=7 (256 DW),
        // pad_amount=3 (4 DW) -> LDS row stride 260 floats. Tensor 256 x 32,
        // tile 256 x 32, dim0 stride 256 elements.
        v8i g1;
        g1[0] = (int)((2u << 16) | (1u << 20) | (7u << 22) | (3u << 25));
        g1[1] = (int)(256u << 16);                   // tensor_dim0[15:0] @ bits 63:48
        g1[2] = (int)(32u << 16);                    // tensor_dim0[31:16]=0 | tensor_dim1[15:0]=32
        g1[3] = (int)(256u << 16);                   // tensor_dim1[31:16]=0 | tile_dim0=256
        g1[4] = (int)32u;                            // tile_dim1=32 | tile_dim2=0
        g1[5] = (int)256u;                           // tensor_dim0_stride[31:0] = 256 elements
        g1[6] = 0;                                   // stride hi | tensor_dim1_stride lo (unused, 2-D)
        g1[7] = 0;
        v4i gz4 = {0, 0, 0, 0};                      // groups 2/3: unused dims for 2-D tile
        v8i gz8 = {0, 0, 0, 0, 0, 0, 0, 0};         // trailing descriptor words (6-arg form), zero

        // 6-arg form (this toolchain): (g0, g1, g2, g3, ext, cpol)
        __builtin_amdgcn_tensor_load_to_lds(g0, g1, gz4, gz4, gz8, 0);
        __builtin_amdgcn_s_wait_tensorcnt(0);        // issuing wave drains TENSORcnt
    }
#else
    // ---- Fallback: manual coalesced b128 staging ----
    const float4* src = (const float4*)(in + (size_t)rowBase * 256);
    for (int it = 0; it < 32; ++it) {
        int li = it * 64 + tid;
        int r  = li >> 6, c4 = li & 63;
        *(float4*)&Ain[r * A_STR + c4 * 4] = src[li];
    }
#endif
    __syncthreads();                                 // publish LDS to both waves

    const int wave = tid >> 5;           // 16-row tile id within WG
    const int lane = tid & 31;
    const int half = lane >> 4;          // K sub-pair select (ISA 16x4 f32 A layout)
    const int nrow = lane & 15;          // A: M row; B/D: N column
    const float* Abase = &Ain[(wave * 16 + nrow) * A_STR + 2 * half];
    const float* Bbase = &Wl[nrow * W_STR + 2 * half];

    v8f acc = {};
#pragma unroll 4
    for (int k0 = 0; k0 < 256; k0 += 4) {
        v2f a = *(const v2f*)(Abase + k0);           // lanes 0-15: K{0,1}, 16-31: K{2,3}
        v2f b = *(const v2f*)(Bbase + k0);           // rows 4..15 read pre-zeroed LDS
        acc = __builtin_amdgcn_wmma_f32_16x16x4_f32(
            /*neg_a=*/false, a, /*neg_b=*/false, b,
            /*c_mod=*/(short)0, acc, /*reuse_a=*/false, /*reuse_b=*/false);
    }

    // D layout: lanes 0-15 N=lane, M=v; lanes 16-31 N=lane-16, M=8+v
    if (nrow < 4) {
        int row0 = rowBase + wave * 16 + half * 8;
#pragma unroll
        for (int v = 0; v < 8; ++v)
            xdot[(size_t)(row0 + v) * 4 + nrow] = acc[v];
    }
}

// ---------------------------------------------------------------------------
// Phase B: serial scan. One lane per batch (128 chains of length 1024).
// lin_k = xdot[t,b,k] + wh_k*h + bias_k;  wh_k = sum_h W[k][256+h]
// ---------------------------------------------------------------------------
__global__ __launch_bounds__(32) void qlstm_scan(const float* __restrict__ xdot,
                                                 const float* __restrict__ W,
                                                 const float* __restrict__ bias,
                                                 float* __restrict__ hvals,
                                                 float* __restrict__ cfin) {
    __shared__ float wh[4], bb[4];
    const int tid = threadIdx.x;
    if (tid < 4) {
        float s = 0.0f;
        for (int j = 0; j < 256; ++j) s += W[tid * 512 + 256 + j];
        wh[tid] = s;
        bb[tid] = bias[tid];
    }
    __syncthreads();

    const int b = blockIdx.x * 32 + tid;
    const float w0 = wh[0], w1 = wh[1], w2 = wh[2], w3 = wh[3];
    const float b0 = bb[0], b1 = bb[1], b2 = bb[2], b3 = bb[3];
    float c = 0.0f, h = 0.0f;

    for (int t = 0; t < T_DIM; ++t) {
        const float4 xd = *(const float4*)&xdot[((size_t)t * B_DIM + b) * 4];
        float q0 = __sinf(fmaf(w0, h, xd.x) + b0);   // v_sin_f32
        float q1 = __sinf(fmaf(w1, h, xd.y) + b1);
        float q2 = __sinf(fmaf(w2, h, xd.z) + b2);
        float q3 = __sinf(fmaf(w3, h, xd.w) + b3);
        float f  = fast_sigmoid(q0);
        float ig = fast_sigmoid(q1) * fast_tanh(q2);
        float o  = fast_sigmoid(q3);
        c = fmaf(f, c, ig);
        h = o * fast_tanh(c);
        hvals[t * B_DIM + b] = h;
    }
    cfin[b] = c;
}

// ---------------------------------------------------------------------------
// Phase C: broadcast scalar h(t,b) over H; append final hx and cx rows.
// Pure streaming store: float4 per thread, ~134 MB @ HBM rate.
// ---------------------------------------------------------------------------
__global__ __launch_bounds__(256) void qlstm_bcast(const float* __restrict__ hvals,
                                                   const float* __restrict__ cfin,
                                                   float4* __restrict__ out, int n4) {
    const int i = blockIdx.x * 256 + threadIdx.x;
    if (i >= n4) return;
    const int TBH4 = ROWS * (H_DIM / 4);       // 8,388,608 float4 (stacked)
    const int BH4  = B_DIM * (H_DIM / 4);      // 8,192 float4 (per [B,H] tensor)
    float v;
    if (i < TBH4)              v = hvals[i >> 6];
    else if (i < TBH4 + BH4)   v = hvals[(T_DIM - 1) * B_DIM + ((i - TBH4) >> 6)];
    else                       v = cfin[(i - TBH4 - BH4) >> 6];
    out[i] = make_float4(v, v, v, v);
}

extern "C" void kernel_launch(void* const* d_in, const int* in_sizes, int n_in,
                              void* d_out, int out_size, void* d_ws, size_t ws_size,
                              hipStream_t stream) {
    const float* in   = (const float*)d_in[0];  // [T,B,D] fp32
    const float* W    = (const float*)d_in[1];  // [4, D+H] fp32
    const float* bias = (const float*)d_in[2];  // [4] fp32
    // d_in[3] (qw) provably does not affect the output (<X> invariant under RX; H maps X->Z)

    float* ws    = (float*)d_ws;
    float* xdot  = ws;                               // ROWS*4      = 524,288 floats (2 MB)
    float* hvals = ws + (size_t)ROWS * 4;            // ROWS        = 131,072 floats (512 KB)
    float* cfin  = hvals + ROWS;                     // B_DIM       = 128 floats

    qlstm_xdot_wmma<<<ROWS / 32, 64, 0, stream>>>(in, W, xdot);
    qlstm_scan<<<B_DIM / 32, 32, 0, stream>>>(xdot, W, bias, hvals, cfin);

    const int n4 = out_size / 4;                     // 8,404,992 float4
    qlstm_bcast<<<(n4 + 255) / 256, 256, 0, stream>>>(hvals, cfin, (float4*)d_out, n4);
}